// Concurrent_13623636263650
// MI455X (gfx1250) — compile-verified
//
#include <hip/hip_runtime.h>
#include <math.h>

// Problem constants (match reference)
#define BB   16
#define NN   2000
#define CC   64
#define DIN  16
#define DOUT 32
#define NT   125      // NN / 16, exact
#define KEXT 272      // 256 (ox⊗xg1) + 16 (ox for bias fold)

typedef __attribute__((ext_vector_type(2))) float v2f;
typedef __attribute__((ext_vector_type(8))) float v8f;

// CDNA5 fp32 WMMA: D(16x16,f32) = A(16x4,f32) * B(4x16,f32) + C
__device__ __forceinline__ v8f wmma4(v2f a, v2f b, v8f c) {
  return __builtin_amdgcn_wmma_f32_16x16x4_f32(
      /*neg_a=*/false, a, /*neg_b=*/false, b,
      /*c_mod=*/(short)0, c, /*reuse_a=*/false, /*reuse_b=*/false);
}

// float <-> totally-ordered uint key (for atomicMin/Max on floats incl. negatives)
__device__ __forceinline__ unsigned fkey(float f) {
  unsigned u = __float_as_uint(f);
  return u ^ ((unsigned)((int)u >> 31) | 0x80000000u);
}
__device__ __forceinline__ float funkey(unsigned k) {
  unsigned u = (k & 0x80000000u) ? (k ^ 0x80000000u) : ~k;
  return __uint_as_float(u);
}

// keys[0..15] = per-batch min key (init 0xFFFFFFFF), keys[16..31] = max key (init 0)
__global__ void k_init(unsigned* keys) {
  int t = threadIdx.x;
  if (t < 32) keys[t] = (t < 16) ? 0xFFFFFFFFu : 0u;
}

// Row-normalize res_x -> y.  grid = BB*NN blocks of 64 threads.
__global__ void k_norm(const float* __restrict__ x, float* __restrict__ y) {
  int row = blockIdx.x;          // b*NN + n
  int c = threadIdx.x;           // 0..63
  float v = x[(size_t)row * CC + c];
  __shared__ float red[64];
  red[c] = v * v;
  __syncthreads();
  for (int s = 32; s > 0; s >>= 1) {
    if (c < s) red[c] += red[c + s];
    __syncthreads();
  }
  y[(size_t)row * CC + c] = v / sqrtf(red[0]);
}

// s[b,c] = sum_m y[b,m,c].  grid = (CC, BB), 256 threads.
__global__ void k_colsum(const float* __restrict__ y, float* __restrict__ sarr) {
  int c = blockIdx.x, b = blockIdx.y, tid = threadIdx.x;
  float acc = 0.f;
  for (int m = tid; m < NN; m += 256) acc += y[((size_t)(b * NN + m)) * CC + c];
  __shared__ float red[256];
  red[tid] = acc;
  __syncthreads();
  for (int s = 128; s > 0; s >>= 1) {
    if (tid < s) red[tid] += red[tid + s];
    __syncthreads();
  }
  if (tid == 0) sarr[b * CC + c] = red[0];
}

// Tiled Gram Y·Yᵀ (upper triangle of 64x64 macro-tiles), diag zeroed,
// block-reduced min/max -> key atomics.  grid = (32*32, BB), 256 threads (8 waves).
__global__ void __launch_bounds__(256) k_gram(const float* __restrict__ y,
                                              unsigned* __restrict__ keys) {
  const int Iblk = blockIdx.x >> 5, Jblk = blockIdx.x & 31;
  if (Iblk > Jblk) return;                 // symmetric: values identical below diagonal
  const int b = blockIdx.y;
  const int tid = threadIdx.x, w = tid >> 5, lane = tid & 31;
  const int ll = lane & 15, koff = 2 * (lane >> 4), hl = lane >> 4;
  const float* yb = y + (size_t)b * NN * CC;

  float vmin = INFINITY, vmax = -INFINITY;
  for (int s = w; s < 16; s += 8) {        // 16 sub-tiles of 16x16 per macro-tile
    const int I = Iblk * 4 + (s >> 2);
    const int J = Jblk * 4 + (s & 3);
    if (I >= NT || J >= NT) continue;
    // A frag: row I*16+ll, K = k+koff..+1 ; B frag identical pattern from row block J
    const float* arow = yb + (size_t)(I * 16 + ll) * CC + koff;
    const float* brow = yb + (size_t)(J * 16 + ll) * CC + koff;
    v8f c = {};
    #pragma unroll
    for (int k = 0; k < CC; k += 4) {
      v2f a = *(const v2f*)(arow + k);
      v2f bf = *(const v2f*)(brow + k);
      c = wmma4(a, bf, c);
    }
    const int gn = J * 16 + ll;
    #pragma unroll
    for (int r = 0; r < 8; r++) {
      int gm = I * 16 + r + 8 * hl;        // C layout: M = r (+8 for lanes 16-31)
      float v = (gm == gn) ? 0.0f : c[r];  // zero the diagonal
      vmin = fminf(vmin, v);
      vmax = fmaxf(vmax, v);
    }
  }
  __shared__ float rmin[256], rmax[256];
  rmin[tid] = vmin; rmax[tid] = vmax;
  __syncthreads();
  for (int s = 128; s > 0; s >>= 1) {
    if (tid < s) {
      rmin[tid] = fminf(rmin[tid], rmin[tid + s]);
      rmax[tid] = fmaxf(rmax[tid], rmax[tid + s]);
    }
    __syncthreads();
  }
  if (tid == 0) {
    atomicMin(&keys[b], fkey(rmin[0]));
    atomicMax(&keys[16 + b], fkey(rmax[0]));
  }
}

// Per node: rawrow = y·s - 1 ; rowsum' = (rawrow - N*gmin)/range ; d = rowsum'^-1/2
// (inf -> 0) ; z = d * origin_x.
__global__ void k_rowd(const float* __restrict__ y, const float* __restrict__ sarr,
                       const float* __restrict__ ox, const unsigned* __restrict__ keys,
                       float* __restrict__ dbuf, float* __restrict__ z) {
  int idx = blockIdx.x * blockDim.x + threadIdx.x;   // b*NN + m
  if (idx >= BB * NN) return;
  int b = idx / NN;
  float gmin = funkey(keys[b]);
  float range = funkey(keys[16 + b]) - gmin;
  const float* yr = y + (size_t)idx * CC;
  const float* sb = sarr + b * CC;
  float raw = -1.0f;                                  // subtract zeroed diagonal (==1)
  for (int c = 0; c < CC; ++c) raw += yr[c] * sb[c];
  float rowsum = (raw - (float)NN * gmin) / range;
  float dv = 1.0f / sqrtf(rowsum);
  if (isinf(dv)) dv = 0.0f;
  dbuf[idx] = dv;
  const float* oxr = ox + (size_t)idx * DIN;
  float* zr = z + (size_t)idx * DIN;
  for (int i = 0; i < DIN; ++i) zr[i] = dv * oxr[i];
}

// p[b,c,i] = sum_m y[b,m,c]*z[b,m,i] for c<64 ; slot c==64 gives t0[b,i] = sum_m z.
// grid = (65, BB), 256 threads.
__global__ void k_pt0(const float* __restrict__ y, const float* __restrict__ z,
                      float* __restrict__ p) {
  int c = blockIdx.x, b = blockIdx.y, tid = threadIdx.x;
  float acc[DIN];
  #pragma unroll
  for (int i = 0; i < DIN; ++i) acc[i] = 0.f;
  for (int m = tid; m < NN; m += 256) {
    float yv = (c < CC) ? y[((size_t)(b * NN + m)) * CC + c] : 1.0f;
    const float* zr = z + ((size_t)(b * NN + m)) * DIN;
    #pragma unroll
    for (int i = 0; i < DIN; ++i) acc[i] += yv * zr[i];
  }
  __shared__ float red[256];
  for (int i = 0; i < DIN; ++i) {
    red[tid] = acc[i];
    __syncthreads();
    for (int s = 128; s > 0; s >>= 1) {
      if (tid < s) red[tid] += red[tid + s];
      __syncthreads();
    }
    if (tid == 0) p[((size_t)(b * 65 + c)) * DIN + i] = red[0];
    __syncthreads();
  }
}

// Final gconv: per 16-node tile, out(16x32) = tanh( A(16x272) · Wflat(272x32) )
// A[n, d*16+i] = ox[n,d]*xg1[n,i] ; A[n, 256+d] = ox[n,d] (bias fold).
// grid = (16, BB), 256 threads (8 waves, one node-tile each).
__global__ void __launch_bounds__(256) k_gconv(
    const float* __restrict__ y, const float* __restrict__ ox,
    const float* __restrict__ Wp, const float* __restrict__ bp,
    const float* __restrict__ dbuf, const float* __restrict__ z,
    const float* __restrict__ p, const unsigned* __restrict__ keys,
    float* __restrict__ out) {
  const int b = blockIdx.y;
  const int tid = threadIdx.x, w = tid >> 5, lane = tid & 31;
  const int t = blockIdx.x * 8 + w;                  // node tile [0, NT)
  const int ll = lane & 15, koff = 2 * (lane >> 4), hl = lane >> 4;

  __shared__ float sWf[KEXT * DOUT];                 // weights_pool ++ bias_pool
  __shared__ float sp[CC * DIN];                     // p for this batch
  __shared__ float st0[DIN];
  __shared__ float sox[8][16][16];
  __shared__ float sxg[8][16][16];

  for (int idx = tid; idx < 256 * DOUT; idx += 256) sWf[idx] = Wp[idx];
  for (int idx = tid; idx < DIN * DOUT; idx += 256) sWf[256 * DOUT + idx] = bp[idx];
  for (int idx = tid; idx < CC * DIN; idx += 256)
    sp[idx] = p[((size_t)(b * 65)) * DIN + idx];
  if (tid < DIN) st0[tid] = p[((size_t)(b * 65 + 64)) * DIN + tid];

  const float gmin = funkey(keys[b]);
  const float range = funkey(keys[16 + b]) - gmin;

  if (t < NT) {
    for (int j = lane; j < 256; j += 32) {
      int n = j >> 4, i = j & 15;
      sox[w][n][i] = ox[((size_t)(b * NN + t * 16 + n)) * DIN + i];
    }
  }
  __syncthreads();

  if (t < NT) {                                      // xg1 = (d/range)*(Y·p - z - gmin*t0)
    for (int j = lane; j < 256; j += 32) {
      int n = j >> 4, i = j & 15;
      int gn = t * 16 + n;
      float acc = -z[((size_t)(b * NN + gn)) * DIN + i];
      const float* yr = y + (size_t)(b * NN + gn) * CC;
      for (int c2 = 0; c2 < CC; ++c2) acc += yr[c2] * sp[c2 * DIN + i];
      acc -= gmin * st0[i];
      sxg[w][n][i] = (dbuf[b * NN + gn] / range) * acc;
    }
  }
  __syncthreads();

  if (t < NT) {
    v8f c0 = {}, c1 = {};
    for (int k = 0; k < 256; k += 4) {               // outer-product part of A
      int kk = k + koff;
      v2f a, b0, b1;
      a.x = sox[w][ll][kk >> 4] * sxg[w][ll][kk & 15];
      a.y = sox[w][ll][(kk + 1) >> 4] * sxg[w][ll][(kk + 1) & 15];
      b0.x = sWf[kk * DOUT + ll];        b0.y = sWf[(kk + 1) * DOUT + ll];
      b1.x = sWf[kk * DOUT + 16 + ll];   b1.y = sWf[(kk + 1) * DOUT + 16 + ll];
      c0 = wmma4(a, b0, c0);
      c1 = wmma4(a, b1, c1);
    }
    for (int k = 256; k < KEXT; k += 4) {            // bias part of A (raw ox)
      int kk = k + koff;
      v2f a, b0, b1;
      a.x = sox[w][ll][kk - 256];
      a.y = sox[w][ll][kk + 1 - 256];
      b0.x = sWf[kk * DOUT + ll];        b0.y = sWf[(kk + 1) * DOUT + ll];
      b1.x = sWf[kk * DOUT + 16 + ll];   b1.y = sWf[(kk + 1) * DOUT + 16 + ll];
      c0 = wmma4(a, b0, c0);
      c1 = wmma4(a, b1, c1);
    }
    #pragma unroll
    for (int r = 0; r < 8; r++) {
      int node = t * 16 + r + 8 * hl;
      out[((size_t)(b * DOUT + ll)) * NN + node]      = tanhf(c0[r]);
      out[((size_t)(b * DOUT + 16 + ll)) * NN + node] = tanhf(c1[r]);
    }
  }
}

extern "C" void kernel_launch(void* const* d_in, const int* in_sizes, int n_in,
                              void* d_out, int out_size, void* d_ws, size_t ws_size,
                              hipStream_t stream) {
  const float* res = (const float*)d_in[0];   // (B,N,C)
  const float* ox  = (const float*)d_in[1];   // (B,N,16)
  const float* Wp  = (const float*)d_in[2];   // (16,16,32) == Wflat rows 0..255
  const float* bp  = (const float*)d_in[3];   // (16,32)    == Wflat rows 256..271
  float* out = (float*)d_out;                 // (B,32,N)

  char* ws = (char*)d_ws;                     // ~10.44 MB total
  float*    y    = (float*)(ws + 0);          // B*N*C        = 8,192,000 B
  float*    sarr = (float*)(ws + 8192000);    // B*C          =     4,096 B
  unsigned* keys = (unsigned*)(ws + 8196096); // 2*B keys     =       128 B
  float*    dbuf = (float*)(ws + 8196224);    // B*N          =   128,000 B
  float*    z    = (float*)(ws + 8324224);    // B*N*16       = 2,048,000 B
  float*    p    = (float*)(ws + 10372224);   // B*65*16      =    66,560 B

  k_init  <<<1, 32, 0, stream>>>(keys);
  k_norm  <<<BB * NN, 64, 0, stream>>>(res, y);
  k_colsum<<<dim3(CC, BB), 256, 0, stream>>>(y, sarr);
  k_gram  <<<dim3(32 * 32, BB), 256, 0, stream>>>(y, keys);
  k_rowd  <<<(BB * NN + 255) / 256, 256, 0, stream>>>(y, sarr, ox, keys, dbuf, z);
  k_pt0   <<<dim3(65, BB), 256, 0, stream>>>(y, z, p);
  k_gconv <<<dim3(16, BB), 256, 0, stream>>>(y, ox, Wp, bp, dbuf, z, p, keys, out);
}